// MetaLayerBP_single_50242527429375
// MI455X (gfx1250) — compile-verified
//
#include <hip/hip_runtime.h>
#include <hip/hip_bf16.h>

// fp32 WMMA fragments for V_WMMA_F32_16X16X4_F32 (wave32):
//   A: 16x4 f32 -> v2f ; B: 4x16 f32 -> v2f ; C/D: 16x16 f32 -> v8f
typedef float v2f __attribute__((ext_vector_type(2)));
typedef float v8f __attribute__((ext_vector_type(8)));

__device__ __forceinline__ v8f wmma_f32(v2f a, v2f b, v8f c) {
    return __builtin_amdgcn_wmma_f32_16x16x4_f32(false, a, false, b, (short)0, c, false, false);
}

__global__ __launch_bounds__(256) void zero_f32_kernel(float* __restrict__ p, int n) {
    for (int i = blockIdx.x * blockDim.x + threadIdx.x; i < n; i += gridDim.x * blockDim.x)
        p[i] = 0.0f;
}

#define EIN_PITCH 100   // 96 + pad  (400B rows: 16B-aligned float4 stores, 8B-aligned v2f loads)
#define NIN_PITCH 132   // 128 + pad (528B rows)
#define H_PITCH   68    // 64 + pad  (272B rows)

// Pair-interleave a row-major [K][N] weight into LDS so a B fragment
// {W[k][c], W[k+1][c]} (k even) is one contiguous 8B element:
//   sWp[(k>>1)*(2N) + c*2 + (k&1)] = W[k*N + c]
template <int K, int N>
__device__ __forceinline__ void stage_weights_pairs(const float* __restrict__ W,
                                                    float* __restrict__ sWp,
                                                    int tid, int nthreads) {
    for (int i = tid; i < K * N; i += nthreads) {
        int k = i / N, c = i % N;
        sWp[(k >> 1) * (2 * N) + c * 2 + (k & 1)] = W[i];
    }
}

// ---------------- Edge model: new_ea = relu([x[row]|x[col]|ea] @ We1 + be1) @ We2 + be2
// ----------------              + scatter: agg[col] += new_ea
// One wave = 32 edges (two 16-row M-tiles sharing every B fragment).
__global__ __launch_bounds__(128)
void edge_kernel(const float* __restrict__ x,
                 const long long* __restrict__ edge_index,
                 const float* __restrict__ edge_attr,
                 const float* __restrict__ We1, const float* __restrict__ be1,
                 const float* __restrict__ We2, const float* __restrict__ be2,
                 float* __restrict__ new_edge_attr,
                 float* __restrict__ agg,
                 int n_edges)
{
    __shared__ float sWe1p[96 * 64];                // 24 KB, pair-interleaved
    __shared__ float sWe2p[64 * 32];                //  8 KB, pair-interleaved
    __shared__ float sEin[4][32 * EIN_PITCH];       // 51.2 KB
    __shared__ float sH[4][32 * H_PITCH];           // 34.8 KB

    const int tid  = threadIdx.x;
    const int wave = tid >> 5;
    const int lane = tid & 31;
    const int m    = lane & 15;
    const int koff = (lane >> 4) * 2;               // K offset 0 / 2 per half-wave
    const int mlo  = (lane >> 4) * 8;               // C-layout: reg j -> M = mlo + j

    stage_weights_pairs<96, 64>(We1, sWe1p, tid, 128);
    stage_weights_pairs<64, 32>(We2, sWe2p, tid, 128);
    __syncthreads();

    const int  tile   = blockIdx.x * 4 + wave;      // 32-edge tile
    const bool active = (tile * 32) < n_edges;
    const long long e0 = active ? (long long)tile * 32 : 0;

    const long long* rowp = edge_index;
    const long long* colp = edge_index + n_edges;

    // ---- stage e_in[32][96] = [x[row] | x[col] | edge_attr] via float4 gathers
    float* ein = sEin[wave];
    #pragma unroll
    for (int it = 0; it < 24; ++it) {
        int i = lane + it * 32;          // 0..767 : 32 edges x 24 float4
        int e = i / 24;
        int q = i % 24;
        long long eg = e0 + e;
        const float* src;
        if (q < 8)       src = x + (size_t)rowp[eg] * 32 + q * 4;
        else if (q < 16) src = x + (size_t)colp[eg] * 32 + (q - 8) * 4;
        else             src = edge_attr + (size_t)eg * 32 + (q - 16) * 4;
        *(float4*)(ein + e * EIN_PITCH + q * 4) = *(const float4*)src;
    }
    __syncthreads();

    // ---- GEMM1: [32,96] @ [96,64]; B fragment shared across both M-tiles
    v8f acc[2][4] = {};
    for (int kk = 0; kk < 96; kk += 4) {
        const int kp = kk + koff;
        v2f a0 = *(const v2f*)(ein + m * EIN_PITCH + kp);
        v2f a1 = *(const v2f*)(ein + (16 + m) * EIN_PITCH + kp);
        const float* wrow = sWe1p + (kp >> 1) * 128;   // pair-row of 64 cols x 2
        #pragma unroll
        for (int n = 0; n < 4; ++n) {
            v2f b = *(const v2f*)(wrow + (n * 16 + m) * 2);
            acc[0][n] = wmma_f32(a0, b, acc[0][n]);
            acc[1][n] = wmma_f32(a1, b, acc[1][n]);
        }
    }

    // ---- bias + relu, restage h[32][64] row-major for GEMM2 A-fragments
    float* hbuf = sH[wave];
    {
        float bb[4] = { be1[m], be1[16 + m], be1[32 + m], be1[48 + m] };
        #pragma unroll
        for (int t = 0; t < 2; ++t)
            #pragma unroll
            for (int j = 0; j < 8; ++j) {
                float* hr = hbuf + (t * 16 + mlo + j) * H_PITCH;
                #pragma unroll
                for (int n = 0; n < 4; ++n) {
                    float h = acc[t][n][j] + bb[n];
                    hr[n * 16 + m] = h > 0.f ? h : 0.f;
                }
            }
    }
    __syncthreads();

    // ---- GEMM2: [32,64] @ [64,32]
    v8f d[2][2] = {};
    for (int kk = 0; kk < 64; kk += 4) {
        const int kp = kk + koff;
        v2f a0 = *(const v2f*)(hbuf + m * H_PITCH + kp);
        v2f a1 = *(const v2f*)(hbuf + (16 + m) * H_PITCH + kp);
        const float* wrow = sWe2p + (kp >> 1) * 64;    // pair-row of 32 cols x 2
        #pragma unroll
        for (int n = 0; n < 2; ++n) {
            v2f b = *(const v2f*)(wrow + (n * 16 + m) * 2);
            d[0][n] = wmma_f32(a0, b, d[0][n]);
            d[1][n] = wmma_f32(a1, b, d[1][n]);
        }
    }

    // ---- bias, store new_edge_attr, atomic scatter into agg[col]
    if (active) {
        float bb0 = be2[m], bb1 = be2[16 + m];
        #pragma unroll
        for (int t = 0; t < 2; ++t)
            #pragma unroll
            for (int j = 0; j < 8; ++j) {
                long long eg = e0 + t * 16 + mlo + j;
                float o0 = d[t][0][j] + bb0;
                float o1 = d[t][1][j] + bb1;
                new_edge_attr[eg * 32 + m]      = o0;
                new_edge_attr[eg * 32 + 16 + m] = o1;
                long long cn = colp[eg];
                atomicAdd(&agg[cn * 32 + m],      o0);
                atomicAdd(&agg[cn * 32 + 16 + m], o1);
            }
    }
}

// ---------------- Node model: new_x = relu([x|x_lstm|z|agg] @ Wn1 + bn1) @ Wn2 + bn2
__global__ __launch_bounds__(128)
void node_kernel(const float* __restrict__ x,
                 const float* __restrict__ x_lstm,
                 const float* __restrict__ z,
                 const float* __restrict__ agg,
                 const float* __restrict__ Wn1, const float* __restrict__ bn1,
                 const float* __restrict__ Wn2, const float* __restrict__ bn2,
                 float* __restrict__ new_x,
                 int n_nodes)
{
    __shared__ float sWn1p[128 * 64];               // 32 KB, pair-interleaved
    __shared__ float sWn2p[64 * 32];                //  8 KB, pair-interleaved
    __shared__ float sNin[4][32 * NIN_PITCH];       // 67.6 KB
    __shared__ float sH[4][32 * H_PITCH];           // 34.8 KB

    const int tid  = threadIdx.x;
    const int wave = tid >> 5;
    const int lane = tid & 31;
    const int m    = lane & 15;
    const int koff = (lane >> 4) * 2;
    const int mlo  = (lane >> 4) * 8;

    stage_weights_pairs<128, 64>(Wn1, sWn1p, tid, 128);
    stage_weights_pairs<64, 32>(Wn2, sWn2p, tid, 128);
    __syncthreads();

    const int  tile   = blockIdx.x * 4 + wave;      // 32-node tile
    const bool active = (tile * 32) < n_nodes;
    const long long n0 = active ? (long long)tile * 32 : 0;

    // ---- stage n_in[32][128] = [x | x_lstm | z | agg] (fully coalesced, e = it, q = lane)
    float* nin = sNin[wave];
    #pragma unroll
    for (int it = 0; it < 32; ++it) {
        int q = lane;                    // float4 index within 128
        long long ng = n0 + it;
        const float* src;
        if (q < 8)       src = x      + (size_t)ng * 32 + q * 4;
        else if (q < 16) src = x_lstm + (size_t)ng * 32 + (q - 8) * 4;
        else if (q < 24) src = z      + (size_t)ng * 32 + (q - 16) * 4;
        else             src = agg    + (size_t)ng * 32 + (q - 24) * 4;
        *(float4*)(nin + it * NIN_PITCH + q * 4) = *(const float4*)src;
    }
    __syncthreads();

    // ---- GEMM1: [32,128] @ [128,64]
    v8f acc[2][4] = {};
    for (int kk = 0; kk < 128; kk += 4) {
        const int kp = kk + koff;
        v2f a0 = *(const v2f*)(nin + m * NIN_PITCH + kp);
        v2f a1 = *(const v2f*)(nin + (16 + m) * NIN_PITCH + kp);
        const float* wrow = sWn1p + (kp >> 1) * 128;
        #pragma unroll
        for (int n = 0; n < 4; ++n) {
            v2f b = *(const v2f*)(wrow + (n * 16 + m) * 2);
            acc[0][n] = wmma_f32(a0, b, acc[0][n]);
            acc[1][n] = wmma_f32(a1, b, acc[1][n]);
        }
    }

    float* hbuf = sH[wave];
    {
        float bb[4] = { bn1[m], bn1[16 + m], bn1[32 + m], bn1[48 + m] };
        #pragma unroll
        for (int t = 0; t < 2; ++t)
            #pragma unroll
            for (int j = 0; j < 8; ++j) {
                float* hr = hbuf + (t * 16 + mlo + j) * H_PITCH;
                #pragma unroll
                for (int n = 0; n < 4; ++n) {
                    float h = acc[t][n][j] + bb[n];
                    hr[n * 16 + m] = h > 0.f ? h : 0.f;
                }
            }
    }
    __syncthreads();

    // ---- GEMM2: [32,64] @ [64,32]
    v8f d[2][2] = {};
    for (int kk = 0; kk < 64; kk += 4) {
        const int kp = kk + koff;
        v2f a0 = *(const v2f*)(hbuf + m * H_PITCH + kp);
        v2f a1 = *(const v2f*)(hbuf + (16 + m) * H_PITCH + kp);
        const float* wrow = sWn2p + (kp >> 1) * 64;
        #pragma unroll
        for (int n = 0; n < 2; ++n) {
            v2f b = *(const v2f*)(wrow + (n * 16 + m) * 2);
            d[0][n] = wmma_f32(a0, b, d[0][n]);
            d[1][n] = wmma_f32(a1, b, d[1][n]);
        }
    }

    if (active) {
        float bb0 = bn2[m], bb1 = bn2[16 + m];
        #pragma unroll
        for (int t = 0; t < 2; ++t)
            #pragma unroll
            for (int j = 0; j < 8; ++j) {
                long long ng = n0 + t * 16 + mlo + j;
                new_x[ng * 32 + m]      = d[t][0][j] + bb0;
                new_x[ng * 32 + 16 + m] = d[t][1][j] + bb1;
            }
    }
}

extern "C" void kernel_launch(void* const* d_in, const int* in_sizes, int n_in,
                              void* d_out, int out_size, void* d_ws, size_t ws_size,
                              hipStream_t stream) {
    const float*     x      = (const float*)d_in[0];
    const float*     x_lstm = (const float*)d_in[1];
    const float*     z      = (const float*)d_in[2];
    const long long* eidx   = (const long long*)d_in[3];   // int64 [2, E]
    const float*     eattr  = (const float*)d_in[4];
    // d_in[5], d_in[6]: agent index lists — unused by the reference computation
    const float* We1 = (const float*)d_in[7];
    const float* be1 = (const float*)d_in[8];
    const float* We2 = (const float*)d_in[9];
    const float* be2 = (const float*)d_in[10];
    const float* Wn1 = (const float*)d_in[11];
    const float* bn1 = (const float*)d_in[12];
    const float* Wn2 = (const float*)d_in[13];
    const float* bn2 = (const float*)d_in[14];

    const int n_nodes = in_sizes[0] / 32;
    const int n_edges = in_sizes[4] / 32;

    float* new_x  = (float*)d_out;                          // [N, 32]
    float* new_ea = (float*)d_out + (size_t)n_nodes * 32;   // [E, 32]
    float* agg    = (float*)d_ws;                           // [N, 32] scratch

    // 1) zero the scatter accumulator (harness poisons d_ws)
    int zn = n_nodes * 32;
    zero_f32_kernel<<<(zn + 256 * 16 - 1) / (256 * 16), 256, 0, stream>>>(agg, zn);

    // 2) edge MLP + scatter-add (4 waves/block, 32 edges/wave)
    int e_tiles  = (n_edges + 31) / 32;
    int e_blocks = (e_tiles + 3) / 4;
    edge_kernel<<<e_blocks, 128, 0, stream>>>(x, eidx, eattr, We1, be1, We2, be2,
                                              new_ea, agg, n_edges);

    // 3) node MLP (stream-ordered after edge kernel)
    int n_tiles  = (n_nodes + 31) / 32;
    int n_blocks = (n_tiles + 3) / 4;
    node_kernel<<<n_blocks, 128, 0, stream>>>(x, x_lstm, z, agg, Wn1, bn1, Wn2, bn2,
                                              new_x, n_nodes);
}